// AngularTerms_2972117368842
// MI455X (gfx1250) — compile-verified
//
#include <hip/hip_runtime.h>

// ---------------------------------------------------------------------------
// AngularTerms (ANI angular symmetry functions) for MI455X / gfx1250.
//
// out[p, a, z] = 2 * fcj0*fcj1 * exp(-eta*(dsh - ShfA[a])^2)
//                  * ((1 + cos(arccos(0.95*cos th) - ShfZ[z]))/2)^Zeta
//
// Bandwidth-bound: 512 MB out + 48 MB in => ~24 us floor @ 23.3 TB/s.
// The 8x8 (a,z) outer product per point runs on the matrix unit:
// V_WMMA_F32_16X16X4_F32 with K=4 packs FOUR points per instruction
// (point k -> row-block k>>1 of A, col-block k&1 of B, zeros elsewhere).
// M = z (factor1), N = a (factor2) so each lane's 8 accumulator VGPRs are
// 8 *consecutive* output floats -> two global_store_b128, fully coalesced.
//
// Zeta==32 fast path (5 squarings) specialized at wave level => the 8-step
// WMMA loop fully unrolls; hot-loop exp uses the hardware transcendental
// (v_exp_f32 via __expf, arg <= 0 so no overflow handling needed).
// ---------------------------------------------------------------------------

typedef float v2f __attribute__((ext_vector_type(2)));
typedef float v4f __attribute__((ext_vector_type(4)));
typedef float v8f __attribute__((ext_vector_type(8)));

#define PI_F 3.14159265358979323846f
#define CUTOFF_F 3.5f

template <bool Z32>
__device__ __forceinline__ void wave_body(
    int waveBase, int lane, float c, float s, float dsh, float w,
    float czv, float szv, float sa, float eta, float zeta,
    float* __restrict__ out)
{
    const int  sub8 = lane & 7;
    const int  grp  = lane >> 3;         // which of 4 points this lane serves
    const bool lo   = lane < 8;          // A row-block 0 owner (points k=0,1)
    const bool hi   = lane >= 24;        // A row-block 1 owner (points k=2,3)
    const int  srcDn = hi ? (lane - 8) : lane;   // f1 of point k=2 lives here
    const int  srcUp = lo ? (lane + 8) : lane;   // f1 of point k=1 lives here

#pragma unroll
    for (int it = 0; it < 8; ++it) {     // 8 WMMA steps x 4 points = 32 points
        const int qb = it * 4 + grp;     // point (0..31) this lane's slot serves
        const float cq = __shfl(c,   qb, 32);
        const float sq = __shfl(s,   qb, 32);
        const float dq = __shfl(dsh, qb, 32);
        const float wq = __shfl(w,   qb, 32);

        // factor1 for (point qb, z = lane&7):
        // cos(angle - ShfZ) = c*cos(ShfZ) + s*sin(ShfZ)   (sin(angle) >= 0)
        float t = 0.5f * (1.0f + fmaf(cq, czv, sq * szv));
        t = fmaxf(t, 0.0f);
        float f1;
        if (Z32) {
            const float t2 = t * t, t4 = t2 * t2, t8 = t4 * t4, t16 = t8 * t8;
            f1 = t16 * t16;              // t^32
        } else {
            f1 = powf(t, zeta);
        }
        // factor2 (pre-scaled by 2*fcj_prod) for (point qb, a = lane&7).
        // Argument is <= 0: hardware v_exp_f32 path is safe and ~2 VALU ops.
        const float dd = dq - sa;
        const float g2 = wq * __expf(-eta * dd * dd);

        // A-matrix (16x4 f32) layout:
        //   VGPR a0: lanes 0-7 = f1[p0][z], lanes 24-31 = f1[p2][z], else 0
        //   VGPR a1: lanes 0-7 = f1[p1][z], lanes 24-31 = f1[p3][z], else 0
        const float f1dn = __shfl(f1, srcDn, 32);
        const float f1up = __shfl(f1, srcUp, 32);
        const float a0 = lo ? f1   : (hi ? f1dn : 0.0f);
        const float a1 = lo ? f1up : (hi ? f1   : 0.0f);
        // B-matrix (4x16 f32) layout:
        //   VGPR b0: lanes 0-7 = g2[p0][a], lanes 16-23 = g2[p2][a], else 0
        //   VGPR b1: lanes 8-15 = g2[p1][a], lanes 24-31 = g2[p3][a], else 0
        const float b0 = ((grp & 1) == 0) ? g2 : 0.0f;
        const float b1 = ((grp & 1) == 1) ? g2 : 0.0f;

        v2f A; A[0] = a0; A[1] = a1;
        v2f B; B[0] = b0; B[1] = b1;
        v8f Cm = {};
        // D[m=z][n=a] = factor1[z] * (2*fcj*factor2)[a]: four 8x8 blocks
        v8f D = __builtin_amdgcn_wmma_f32_16x16x4_f32(
            false, A, false, B, (short)0, Cm, false, false);

        // Lane (grp, a=lane&7) owns 8 consecutive floats of point
        // (waveBase + it*4 + grp): out[p*64 + a*8 + z], z = VGPR index.
        float* o = out + (size_t)(waveBase + it * 4 + grp) * 64 + sub8 * 8;
        v4f dLo, dHi;
        dLo[0] = D[0]; dLo[1] = D[1]; dLo[2] = D[2]; dLo[3] = D[3];
        dHi[0] = D[4]; dHi[1] = D[5]; dHi[2] = D[6]; dHi[3] = D[7];
        *(v4f*)o       = dLo;            // global_store_b128
        *(v4f*)(o + 4) = dHi;            // global_store_b128
    }
}

__global__ __launch_bounds__(256) void angular_terms_kernel(
    const float* __restrict__ vec,    // [2, P, 3]
    const float* __restrict__ etaAp,  // [1]
    const float* __restrict__ zetaP,  // [1]
    const float* __restrict__ shfA,   // [8]
    const float* __restrict__ shfZ,   // [8]
    float* __restrict__ out,          // [P, 64]
    int P)
{
    const int lane     = threadIdx.x & 31;
    const int waveBase = blockIdx.x * 256 + (threadIdx.x & ~31);
    if (waveBase >= P) return;

    const float eta  = etaAp[0];
    const float zeta = zetaP[0];
    const bool  z32  = (zeta == 32.0f);

    const int   sub8 = lane & 7;
    const float sa   = shfA[sub8];          // ShfA for this lane's 'a' slot
    const float shz  = shfZ[sub8];          // ShfZ for this lane's 'z' slot
    const float czv  = cosf(shz);
    const float szv  = sinf(shz);

    // ---- per-point scalars: one point per lane (32 points per wave) ----
    const int p = waveBase + lane;
    const size_t pc = (size_t)(p < P ? p : P - 1);
    const float* va = vec + pc * 3;
    const float* vb = vec + (size_t)P * 3 + pc * 3;
    const float x0 = va[0], y0 = va[1], z0 = va[2];
    const float x1 = vb[0], y1 = vb[1], z1 = vb[2];

    const float d0  = sqrtf(fmaf(x0, x0, fmaf(y0, y0, z0 * z0)));
    const float d1  = sqrtf(fmaf(x1, x1, fmaf(y1, y1, z1 * z1)));
    const float dot = fmaf(x0, x1, fmaf(y0, y1, z0 * z1));
    const float c   = 0.95f * dot / fmaxf(d0 * d1, 1e-10f);   // cos(angle)
    const float s   = sqrtf(fmaxf(1.0f - c * c, 0.0f));       // sin(angle) >= 0
    const float fc0 = fmaf(0.5f, cosf(d0 * (PI_F / CUTOFF_F)), 0.5f);
    const float fc1 = fmaf(0.5f, cosf(d1 * (PI_F / CUTOFF_F)), 0.5f);
    const float w   = 2.0f * fc0 * fc1;                       // 2*fcj_prod
    const float dsh = 0.5f * (d0 + d1);                       // dsum_half

    if (waveBase + 32 <= P) {
        // ---- main path: full wave, EXEC all ones (WMMA requirement) ----
        if (z32) {
            wave_body<true >(waveBase, lane, c, s, dsh, w, czv, szv, sa, eta, zeta, out);
        } else {
            wave_body<false>(waveBase, lane, c, s, dsh, w, czv, szv, sa, eta, zeta, out);
        }
    } else {
        // ---- tail: scalar fallback (partial wave; no WMMA, EXEC-safe) ----
        if (p < P) {
            float* o = out + (size_t)p * 64;
            for (int a8 = 0; a8 < 8; ++a8) {
                const float dd = dsh - shfA[a8];
                const float g2 = w * __expf(-eta * dd * dd);
                for (int zz = 0; zz < 8; ++zz) {
                    const float cs = cosf(shfZ[zz]);
                    const float sn = sinf(shfZ[zz]);
                    float t = fmaxf(0.5f * (1.0f + fmaf(c, cs, s * sn)), 0.0f);
                    float f1;
                    if (z32) {
                        const float t2 = t*t, t4 = t2*t2, t8 = t4*t4, t16 = t8*t8;
                        f1 = t16 * t16;
                    } else {
                        f1 = powf(t, zeta);
                    }
                    o[a8 * 8 + zz] = f1 * g2;
                }
            }
        }
    }
}

extern "C" void kernel_launch(void* const* d_in, const int* in_sizes, int n_in,
                              void* d_out, int out_size, void* d_ws, size_t ws_size,
                              hipStream_t stream) {
    const float* vec  = (const float*)d_in[0];   // vectors12 [2,P,3]
    const float* etaA = (const float*)d_in[1];   // [1]
    const float* zeta = (const float*)d_in[2];   // [1]
    const float* shfA = (const float*)d_in[3];   // [8]
    const float* shfZ = (const float*)d_in[4];   // [8]
    float* out = (float*)d_out;                  // [P, 64]

    const int P = in_sizes[0] / 6;               // 2*P*3 flat elements
    const int blocks = (P + 255) / 256;          // 256 threads = 8 wave32
    angular_terms_kernel<<<dim3(blocks), dim3(256), 0, stream>>>(
        vec, etaA, zeta, shfA, shfZ, out, P);
}